// SSM_43739946943085
// MI455X (gfx1250) — compile-verified
//
#include <hip/hip_runtime.h>
#include <hip/hip_bf16.h>

typedef __attribute__((ext_vector_type(2))) float v2f;
typedef __attribute__((ext_vector_type(8))) float v8f;

#define D_MODEL 256
#define S_SIZE  16
#define ROWS    32    // token rows per block (two 16-row WMMA tiles)
#define XPAD    260   // x LDS row stride (floats): 260 % 64 = 4 -> conflict-free b64 fetch
#define BPAD    20    // Bt LDS row stride (floats): 80 bytes, 16B aligned -> b128 row reads

#define LOG2E 1.4426950408889634f
#define LN2   0.6931471805599453f

__global__ __launch_bounds__(512) void ssm_fused_kernel(
    const float* __restrict__ x,    // [8192, 256] flattened (B,M,L,D)
    const float* __restrict__ W1,   // [256, 256]
    const float* __restrict__ b1,   // [256]
    const float* __restrict__ W2,   // [16, 256]
    const float* __restrict__ b2,   // [16]
    const float* __restrict__ A,    // [256, 16]
    float* __restrict__ out)        // [8192, 256]
{
    __shared__ __align__(16) float xs[ROWS * XPAD];   // x strip, padded
    __shared__ __align__(16) float bts[ROWS * BPAD];  // Bt tile [row][s], padded

    const int tid     = threadIdx.x;
    const int lane    = tid & 31;
    const int wave    = tid >> 5;           // 0..15
    const int hrow    = lane & 15;          // row-within-tile (A) / col-within-tile (B,C,D)
    const int koff    = (lane >> 4) * 2;    // K sub-offset: 0 (lanes 0-15) or 2 (lanes 16-31)
    const int mBase   = (lane >> 4) * 8;    // C/D row offset for upper half-wave
    const int rowBase = blockIdx.x * ROWS;  // token row of this block's strip
    const int colBase = wave * 16;          // output-column tile of this wave

    // ---- Seed Bt tile with bias b2 (partials from ds_add_f32 accumulate on top) ----
    {
        int m = tid >> 4, s = tid & 15;     // 512 threads cover 32x16 exactly
        bts[m * BPAD + s] = b2[s];
    }

    // ---- Stage x strip (32 x 256 f32 = 32KB) via async global->LDS b128 copies ----
    {
        const float* gsrc = x + (size_t)rowBase * D_MODEL;
        #pragma unroll
        for (int it = 0; it < 4; ++it) {
            int v = tid + it * 512;          // float4 index 0..2047
            int r = v >> 6;                  // 64 float4 per row
            int c = (v & 63) << 2;
            unsigned int loff = (unsigned int)(unsigned long long)(&xs[r * XPAD + c]);
            const float* g = gsrc + (size_t)v * 4;
            asm volatile("global_load_async_to_lds_b128 %0, %1, off"
                         :: "v"(loff), "v"(g) : "memory");
        }
        asm volatile("s_wait_asynccnt 0x0" ::: "memory");
    }
    __syncthreads();

    // A-operand bases: lane reads x[row][4k + koff + {0,1}] as one b64 per step
    const v2f* ap0 = reinterpret_cast<const v2f*>(&xs[hrow * XPAD + koff]);         // rows 0-15
    const v2f* ap1 = reinterpret_cast<const v2f*>(&xs[(16 + hrow) * XPAD + koff]);  // rows 16-31

    // ---- Waves 0-7: partial Bt tiles = x_strip * W2^T (2 row tiles x 4 K-slices) ----
    if (wave < 8) {
        const int t  = wave >> 2;            // row tile 0 or 1
        const int k0 = (wave & 3) * 16;      // K-step offset of this wave's slice
        const v2f* apt = t ? ap1 : ap0;
        const v2f* sp  = reinterpret_cast<const v2f*>(W2 + (size_t)hrow * D_MODEL + koff);
        v8f cb = {};
        #pragma unroll 8
        for (int k = 0; k < 16; ++k) {
            v2f a = apt[2 * (k0 + k)];
            v2f b = sp[2 * (k0 + k)];
            cb = __builtin_amdgcn_wmma_f32_16x16x4_f32(
                false, a, false, b, (short)0, cb, false, false);
        }
        #pragma unroll
        for (int v = 0; v < 8; ++v) {
            int m = t * 16 + v + mBase;                 // strip row (token)
            atomicAdd(&bts[m * BPAD + hrow], cb[v]);    // hrow == s index; ds_add_f32
        }
    }

    // ---- All waves: two Dp tiles = x_strip * W1^T, B-operand reused across row tiles ----
    const v2f* bp = reinterpret_cast<const v2f*>(
        W1 + (size_t)(colBase + hrow) * D_MODEL + koff);
    v8f c0 = {}, c1 = {};
    #pragma unroll 8
    for (int k = 0; k < 64; ++k) {
        v2f b  = bp[2 * k];                  // one global b64 serves both WMMAs
        v2f a0 = ap0[2 * k];
        v2f a1 = ap1[2 * k];
        c0 = __builtin_amdgcn_wmma_f32_16x16x4_f32(
            false, a0, false, b, (short)0, c0, false, false);
        c1 = __builtin_amdgcn_wmma_f32_16x16x4_f32(
            false, a1, false, b, (short)0, c1, false, false);
    }

    __syncthreads();   // Bt tiles (bias + all partials) visible to all waves

    // ---- Epilogue: per lane, 16 elements (2 tiles x rows mBase..mBase+7, col dcol) ----
    const int dcol = colBase + hrow;
    float a_s[S_SIZE];   // A row for this column, pre-scaled by log2(e) for v_exp_f32
    {
        const float4* Ag = reinterpret_cast<const float4*>(A + (size_t)dcol * S_SIZE);
        #pragma unroll
        for (int q = 0; q < 4; ++q) {
            float4 t = Ag[q];
            a_s[4 * q + 0] = t.x * LOG2E; a_s[4 * q + 1] = t.y * LOG2E;
            a_s[4 * q + 2] = t.z * LOG2E; a_s[4 * q + 3] = t.w * LOG2E;
        }
    }
    const float b1v = b1[dcol];

    #pragma unroll
    for (int t = 0; t < 2; ++t) {
        const v8f& c = t ? c1 : c0;
        #pragma unroll
        for (int v = 0; v < 8; ++v) {
            int   m  = t * 16 + v + mBase;
            float dp = c[v] + b1v;
            float xv = xs[m * XPAD + dcol];

            // softplus(dp) = ln(1 + e^dp); exp2/log2 domain (v_exp_f32 / v_log_f32)
            float et    = __builtin_amdgcn_exp2f(dp * LOG2E);
            float delta = (dp > 20.0f) ? dp : __builtin_amdgcn_logf(1.0f + et) * LN2;

            // Bt row m as 4x ds_load_b128 (half-wave broadcast, conflict-free)
            const float4* btr = reinterpret_cast<const float4*>(&bts[m * BPAD]);
            float4 q0 = btr[0], q1 = btr[1], q2 = btr[2], q3 = btr[3];
            float bt[S_SIZE] = { q0.x, q0.y, q0.z, q0.w,  q1.x, q1.y, q1.z, q1.w,
                                 q2.x, q2.y, q2.z, q2.w,  q3.x, q3.y, q3.z, q3.w };

            float acc = 0.0f, bsq = 0.0f;
            #pragma unroll
            for (int s = 0; s < S_SIZE; ++s) {
                acc += bt[s] * __builtin_amdgcn_exp2f(delta * a_s[s]);  // exp(delta*A[d,s])
                bsq += bt[s] * bt[s];
            }
            float y = acc + xv * delta * bsq;

            // silu(y) = y / (1 + e^-y)
            float ey = __builtin_amdgcn_exp2f(-y * LOG2E);
            y = y * __builtin_amdgcn_rcpf(1.0f + ey);

            out[(size_t)(rowBase + m) * D_MODEL + dcol] = xv * dp + y;
        }
    }
}

extern "C" void kernel_launch(void* const* d_in, const int* in_sizes, int n_in,
                              void* d_out, int out_size, void* d_ws, size_t ws_size,
                              hipStream_t stream) {
    const float* x  = (const float*)d_in[0];
    const float* W1 = (const float*)d_in[1];
    const float* b1 = (const float*)d_in[2];
    const float* W2 = (const float*)d_in[3];
    const float* b2 = (const float*)d_in[4];
    const float* A  = (const float*)d_in[5];
    float* out = (float*)d_out;

    // 8192 token rows / 32 per block = 256 blocks; 512 threads = 16 wave32
    ssm_fused_kernel<<<256, 512, 0, stream>>>(x, W1, b1, W2, b2, A, out);
}